// Filter_61692910240141
// MI455X (gfx1250) — compile-verified
//
#include <hip/hip_runtime.h>
#include <stdint.h>

// ---------------- problem constants (match reference) ----------------
#define BB 4
#define NN 8192
#define CC 80
#define MAX_DET 300
#define NMS_THR 0.5f
#define SCORE_THR 0.05f
#define NW 256          // words per bitmask row = NN/32

typedef __attribute__((ext_vector_type(4))) unsigned int v4u;
typedef __attribute__((ext_vector_type(8))) int v8i;
typedef __attribute__((ext_vector_type(4))) int v4i;

#ifndef __has_builtin
#define __has_builtin(x) 0
#endif
#if __has_builtin(__builtin_amdgcn_tensor_load_to_lds)
#define HAVE_TDM 1
#else
#define HAVE_TDM 0
#endif

#if HAVE_TDM
// Tensor Data Mover: load a 2D f32 tile (tile_d0 elems/row, tile_d1 rows,
// row stride = stride_elems) from global into LDS at byte offset lds_addr.
// D# layout per CDNA5 ISA ch.8 (group0 128b, group1 256b, groups2/3 zero => <=2D).
// This toolchain's builtin is the 6-arg form:
//   (uint32x4 g0, int32x8 g1, int32x4 g2, int32x4 g3, int32x8 pad, i32 cpol)
__device__ __forceinline__ void tdm_load_f32(unsigned lds_addr,
                                             const float* gptr,
                                             unsigned tile_d0, unsigned tile_d1,
                                             unsigned stride_elems) {
  unsigned long long ga = (unsigned long long)(const void*)gptr;
  v4u g0;
  g0.x = 1u;                                      // count=1 (valid user descriptor)
  g0.y = lds_addr;                                // lds_addr  bits [63:32]
  g0.z = (unsigned)(ga & 0xFFFFFFFFull);          // global_addr bits [95:64]
  g0.w = (unsigned)((ga >> 32) & 0x1FFFFFFull)    // global_addr bits [120:96]
       | (2u << 30);                              // type=2 ("image") bits [127:126]
  v8i g1;
  g1[0] = (int)(2u << 16);                                  // data_size=2 -> 4 bytes
  g1[1] = (int)((tile_d0 & 0xFFFFu) << 16);                 // tensor_dim0[15:0]
  g1[2] = (int)((tile_d0 >> 16) | ((tile_d1 & 0xFFFFu) << 16)); // dim0[31:16]|dim1[15:0]
  g1[3] = (int)((tile_d1 >> 16) | ((tile_d0 & 0xFFFFu) << 16)); // dim1[31:16]|tile_dim0
  g1[4] = (int)(tile_d1 & 0xFFFFu);                         // tile_dim1 (tile_dim2=0)
  g1[5] = (int)stride_elems;                                // tensor_dim0_stride[31:0]
  g1[6] = 0;
  g1[7] = 0;
  v4i z4; z4[0] = 0; z4[1] = 0; z4[2] = 0; z4[3] = 0;       // D# groups 2/3 (<=2D)
  v8i z8; z8[0] = 0; z8[1] = 0; z8[2] = 0; z8[3] = 0;
  z8[4] = 0; z8[5] = 0; z8[6] = 0; z8[7] = 0;
  __builtin_amdgcn_tensor_load_to_lds(g0, g1, z4, z4, z8, 0);
}
#endif

// ------------- kernel A (first in module): IoU suppression bitmask -------------
// TDM bulk-loads the whole sorted box SoA (128KB) into LDS once per block,
// then each thread owns one row i and emits 256 words of (iou>thr && j>i).
__global__ void __launch_bounds__(256) mask_kernel(const float* __restrict__ planes,
                                                   unsigned* __restrict__ mask) {
  extern __shared__ float lds[];   // 4*NN floats: x1|y1|x2|y2 contiguous
  const int tid = threadIdx.x;
#if HAVE_TDM
  if (tid < 32) {                  // wave 0 issues one TDM: tile 8192 x 4, stride 8192
    tdm_load_f32((unsigned)(unsigned long long)(void*)lds, planes, NN, 4, NN);
    __builtin_amdgcn_s_wait_tensorcnt(0);
  }
#else
  for (int k = tid; k < 4 * NN; k += 256) lds[k] = planes[k];
#endif
  __syncthreads();

  const float* x1 = lds;
  const float* y1 = lds + NN;
  const float* x2 = lds + 2 * NN;
  const float* y2 = lds + 3 * NN;

  int i = blockIdx.x * 256 + tid;
  float xi1 = x1[i], yi1 = y1[i], xi2 = x2[i], yi2 = y2[i];
  float ai = (xi2 - xi1) * (yi2 - yi1);
  unsigned* row = mask + (size_t)i * NW;

  for (int w = 0; w < NW; ++w) {
    unsigned m = 0;
    int j0 = w * 32;
    if (j0 + 31 > i) {            // whole word <= i  -> all zero
#pragma unroll 8
      for (int jj = 0; jj < 32; ++jj) {
        int j = j0 + jj;
        float iw = fminf(xi2, x2[j]) - fmaxf(xi1, x1[j]);
        float ih = fminf(yi2, y2[j]) - fmaxf(yi1, y1[j]);
        iw = fmaxf(iw, 0.0f);
        ih = fmaxf(ih, 0.0f);
        float inter = iw * ih;
        float aj = (x2[j] - x1[j]) * (y2[j] - y1[j]);
        // iou > thr  <=>  inter > thr*(ai+aj-inter+eps)   (no division)
        bool sup = (j > i) && (inter > NMS_THR * (ai + aj - inter + 1e-9f));
        m |= (sup ? 1u : 0u) << jj;
      }
    }
    row[w] = m;
  }
}

// ------------- kernel B: greedy bitmask scan + top-300 compaction -------------
__global__ void __launch_bounds__(256) scan_kernel(const float* __restrict__ planes,
                                                   const int* __restrict__ slab,
                                                   const unsigned* __restrict__ mask,
                                                   float* __restrict__ out, int b) {
  extern __shared__ float lds2[];
  float* sc = lds2;                               // NN floats (sorted scores)
  unsigned* keepw = (unsigned*)(lds2 + NN);       // NW words
  unsigned* pref = keepw + NW;                    // NW words
  const int tid = threadIdx.x;

#if HAVE_TDM
  if (tid < 32) {                                 // TDM the score plane (32KB)
    tdm_load_f32((unsigned)(unsigned long long)(void*)sc, planes + 4 * NN, NN, 1, NN);
    __builtin_amdgcn_s_wait_tensorcnt(0);
  }
#else
  for (int k = tid; k < NN; k += 256) sc[k] = planes[4 * NN + k];
#endif
  __syncthreads();

  // keep init: valid = score > SCORE_THR; thread t owns word t
  unsigned w0 = 0;
  for (int jj = 0; jj < 32; ++jj)
    if (sc[tid * 32 + jj] > SCORE_THR) w0 |= 1u << jj;
  keepw[tid] = w0;
  __syncthreads();

  // greedy scan: for i: if keep[i]: keep &= ~mask_row[i]
  for (int i = 0; i < NN; ++i) {
    if (i + 1 < NN)
      __builtin_prefetch(&mask[(size_t)(i + 1) * NW + tid], 0, 1);
    unsigned kw = keepw[i >> 5];                  // row i never clears bit i
    if ((kw >> (i & 31)) & 1u) {
      keepw[tid] &= ~mask[(size_t)i * NW + tid];
    }
    __syncthreads();
  }

  // prefix-scan of popcounts for rank compaction
  unsigned cnt = __popc(keepw[tid]);
  pref[tid] = cnt;
  __syncthreads();
  for (int off = 1; off < NW; off <<= 1) {
    unsigned v = (tid >= off) ? pref[tid - off] : 0u;
    __syncthreads();
    pref[tid] += v;
    __syncthreads();
  }
  unsigned total = pref[NW - 1];
  unsigned base = pref[tid] - cnt;

  float* ob = out;                                // (B,300,4)
  float* os = out + BB * MAX_DET * 4;             // (B,300,1)
  float* ol = os + BB * MAX_DET;                  // (B,300,1)

  // pad ranks [total, 300) with -1 (disjoint from kept ranks — no race)
  for (unsigned r = total + (unsigned)tid; r < MAX_DET; r += 256) {
    float* bp = ob + ((size_t)b * MAX_DET + r) * 4;
    bp[0] = bp[1] = bp[2] = bp[3] = -1.0f;
    os[b * MAX_DET + r] = -1.0f;
    ol[b * MAX_DET + r] = -1.0f;
  }

  // emit kept detections in score-descending (index-ascending) order
  const float* x1 = planes;
  const float* y1 = planes + NN;
  const float* x2 = planes + 2 * NN;
  const float* y2 = planes + 3 * NN;
  unsigned w = keepw[tid];
  unsigned r = base;
  while (w) {
    int bit = __ffs(w) - 1;
    w &= w - 1;
    if (r < MAX_DET) {
      int i = tid * 32 + bit;
      float* bp = ob + ((size_t)b * MAX_DET + r) * 4;
      bp[0] = x1[i];
      bp[1] = y1[i];
      bp[2] = x2[i];
      bp[3] = y2[i];
      os[b * MAX_DET + r] = sc[i];
      ol[b * MAX_DET + r] = (float)slab[i];
    }
    ++r;
  }
}

// ------------- kernel C: per-batch stable descending bitonic sort -------------
// One workgroup per batch; 8192 (score,idx) pairs live in 64KB LDS.
// Total order: (score desc, idx asc)  == stable argsort(-scores).
__global__ void __launch_bounds__(1024) sort_kernel(const float* __restrict__ scores,
                                                    const int* __restrict__ labels,
                                                    const float* __restrict__ box,
                                                    float* __restrict__ planes_all,
                                                    int* __restrict__ slab_all) {
  extern __shared__ float lds3[];
  float* skey = lds3;                 // NN floats
  int* sval = (int*)(lds3 + NN);      // NN ints
  const int tid = threadIdx.x;
  const int b = blockIdx.x;

  for (int k = tid; k < NN; k += 1024) {
    skey[k] = scores[b * NN + k];
    sval[k] = k;
  }
  __syncthreads();

  for (unsigned k2 = 2; k2 <= NN; k2 <<= 1) {
    for (unsigned j = k2 >> 1; j > 0; j >>= 1) {
      for (unsigned e = tid; e < NN; e += 1024) {
        unsigned ix = e ^ j;
        if (ix > e) {
          float ka = skey[e], kb = skey[ix];
          int va = sval[e], vb = sval[ix];
          bool b_bef_a = (kb > ka) || ((kb == ka) && (vb < va));
          bool a_bef_b = (ka > kb) || ((ka == kb) && (va < vb));
          bool dirAsc = ((e & k2) == 0);
          if (dirAsc ? b_bef_a : a_bef_b) {
            skey[e] = kb; skey[ix] = ka;
            sval[e] = vb; sval[ix] = va;
          }
        }
      }
      __syncthreads();
    }
  }

  // gather into SoA planes [x1|y1|x2|y2|score] for TDM-friendly layout
  float* pb = planes_all + (size_t)b * 5 * NN;
  for (int k = tid; k < NN; k += 1024) {
    int o = sval[k];
    const float* bx = box + ((size_t)b * NN + o) * 4;
    pb[0 * NN + k] = bx[0];
    pb[1 * NN + k] = bx[1];
    pb[2 * NN + k] = bx[2];
    pb[3 * NN + k] = bx[3];
    pb[4 * NN + k] = skey[k];
    slab_all[b * NN + k] = labels[b * NN + o];
  }
}

// ------------- kernel D: per-box max/argmax over 80 classes -------------
__global__ void score_kernel(const float* __restrict__ cls,
                             float* __restrict__ scores,
                             int* __restrict__ labels) {
  int t = blockIdx.x * blockDim.x + threadIdx.x;     // 0 .. BB*NN-1
  if (t >= BB * NN) return;
  const float* c = cls + (size_t)t * CC;
  float best = c[0];
  int bi = 0;
  for (int k = 1; k < CC; ++k) {
    float v = c[k];
    if (v > best) { best = v; bi = k; }              // first-max == jnp.argmax
  }
  scores[t] = best;
  labels[t] = bi;
}

// ---------------------------- launcher ----------------------------
extern "C" void kernel_launch(void* const* d_in, const int* in_sizes, int n_in,
                              void* d_out, int out_size, void* d_ws, size_t ws_size,
                              hipStream_t stream) {
  (void)in_sizes; (void)n_in; (void)out_size; (void)ws_size;
  const float* box = (const float*)d_in[0];   // (B,N,4)
  const float* cls = (const float*)d_in[1];   // (B,N,C)
  float* out = (float*)d_out;

  float* wsf = (float*)d_ws;
  const int BN = BB * NN;
  float* scores = wsf;                                   // BN floats
  int* labels = (int*)(wsf + BN);                        // BN ints
  float* planes = wsf + 2 * BN;                          // BB*5*NN floats (SoA)
  int* slab = (int*)(wsf + 2 * BN + 5 * BN);             // BN ints
  unsigned* mask = (unsigned*)(wsf + 2 * BN + 6 * BN);   // NN*NW words (reused per batch)

  score_kernel<<<(BN + 255) / 256, 256, 0, stream>>>(cls, scores, labels);
  sort_kernel<<<BB, 1024, (size_t)NN * 8, stream>>>(scores, labels, box, planes, slab);
  for (int b = 0; b < BB; ++b) {
    mask_kernel<<<NN / 256, 256, (size_t)4 * NN * 4, stream>>>(
        planes + (size_t)b * 5 * NN, mask);
    scan_kernel<<<1, 256, (size_t)(NN * 4 + NW * 8), stream>>>(
        planes + (size_t)b * 5 * NN, slab + b * NN, mask, out, b);
  }
}